// APPNP_68659347194334
// MI455X (gfx1250) — compile-verified
//
#include <hip/hip_runtime.h>

#define N_NODES 100000
#define NFEAT   512
#define NHID    256
#define NCLASS  40
#define N_EDGES 1600000
#define K_HOPS  10

typedef __attribute__((ext_vector_type(2))) float v2f;
typedef __attribute__((ext_vector_type(8))) float v8f;

#define WMMA4(A, B, C) \
  __builtin_amdgcn_wmma_f32_16x16x4_f32(false, (A), false, (B), (short)0, (C), false, false)

// ---------------------------------------------------------------------------
// Fused MLP: h = relu(x @ W1 + b1) @ W2 + b2, one 32-row strip per block.
// 8 waves: wm = wave&1 (16-row half), wn = wave>>1 (64-col group).
// ---------------------------------------------------------------------------
__global__ __launch_bounds__(256) void appnp_mlp(
    const float* __restrict__ x,  const float* __restrict__ W1,
    const float* __restrict__ b1, const float* __restrict__ W2,
    const float* __restrict__ b2, float* __restrict__ h)
{
  constexpr int BM = 32, KC = 32;
  __shared__ float sx[BM][36];    // x tile (stride 36: 16B rows, conflict-free A reads)
  __shared__ float sw[KC][260];   // W1 k-chunk; reused as relu(h1) tile [32][260]

  const int tid   = threadIdx.x;
  const int wave  = tid >> 5;
  const int lane  = tid & 31;
  const int row16 = lane & 15;
  const int kh    = lane >> 4;          // 0: lanes 0-15, 1: lanes 16-31
  const int wm    = wave & 1;
  const int wn    = wave >> 1;
  const int m0    = blockIdx.x * BM;

  v8f acc0 = {}, acc1 = {}, acc2 = {}, acc3 = {};

  for (int kc = 0; kc < NFEAT; kc += KC) {
    __syncthreads();
    { // stage x tile: 32x32 floats = 256 float4, one per thread
      const int r = tid >> 3, c4 = tid & 7;
      const float4 v = *(const float4*)(x + (size_t)(m0 + r) * NFEAT + kc + c4 * 4);
      *(float4*)&sx[r][c4 * 4] = v;
    }
#pragma unroll
    for (int i = 0; i < 8; ++i) { // stage W1 tile: 32x256 floats = 2048 float4
      const int idx = tid + i * 256;
      const int r = idx >> 6, c4 = idx & 63;
      const float4 v = *(const float4*)(W1 + (size_t)(kc + r) * NHID + c4 * 4);
      *(float4*)&sw[r][c4 * 4] = v;
    }
    __syncthreads();
#pragma unroll
    for (int k = 0; k < KC; k += 4) {
      const int krow = k + 2 * kh;                 // A: VGPR0=K(2kh), VGPR1=K(2kh+1)
      v2f a;
      a.x = sx[wm * 16 + row16][krow];
      a.y = sx[wm * 16 + row16][krow + 1];
      const int nb = wn * 64 + row16;              // B: row K striped across lanes
      v2f b0, b1v, b2v, b3v;
      b0.x  = sw[krow][nb];      b0.y  = sw[krow + 1][nb];
      b1v.x = sw[krow][nb + 16]; b1v.y = sw[krow + 1][nb + 16];
      b2v.x = sw[krow][nb + 32]; b2v.y = sw[krow + 1][nb + 32];
      b3v.x = sw[krow][nb + 48]; b3v.y = sw[krow + 1][nb + 48];
      acc0 = WMMA4(a, b0,  acc0);
      acc1 = WMMA4(a, b1v, acc1);
      acc2 = WMMA4(a, b2v, acc2);
      acc3 = WMMA4(a, b3v, acc3);
    }
  }
  __syncthreads();
  { // bias + relu; park h1 tile back into sw (exactly 32x256 used)
    const int nb = wn * 64 + row16;
    const float bb0 = b1[nb], bb1 = b1[nb + 16], bb2 = b1[nb + 32], bb3 = b1[nb + 48];
#pragma unroll
    for (int v = 0; v < 8; ++v) {                  // C/D: VGPR v -> M = v + 8*kh
      const int mr = wm * 16 + 8 * kh + v;
      sw[mr][nb]      = fmaxf(acc0[v] + bb0, 0.0f);
      sw[mr][nb + 16] = fmaxf(acc1[v] + bb1, 0.0f);
      sw[mr][nb + 32] = fmaxf(acc2[v] + bb2, 0.0f);
      sw[mr][nb + 48] = fmaxf(acc3[v] + bb3, 0.0f);
    }
  }
  __syncthreads();
  // GEMM2: h1[32x256] @ W2[256x40] (+b2). NCLASS padded to 64; lane-level guards
  // use selects (cndmask), never diverging around v_wmma (EXEC must stay all-1s).
  v8f acc = {};
  const int  ncol = wn * 16 + row16;    // 0..63
  const bool nval = ncol < NCLASS;
#pragma unroll 4
  for (int k = 0; k < NHID; k += 4) {
    const int krow = k + 2 * kh;
    v2f a;
    a.x = sw[wm * 16 + row16][krow];
    a.y = sw[wm * 16 + row16][krow + 1];
    v2f b;
    b.x = nval ? W2[(size_t)krow * NCLASS + ncol] : 0.0f;
    b.y = nval ? W2[(size_t)(krow + 1) * NCLASS + ncol] : 0.0f;
    acc = WMMA4(a, b, acc);
  }
  if (nval) {
    const float bb = b2[ncol];
#pragma unroll
    for (int v = 0; v < 8; ++v) {
      const int mr = m0 + wm * 16 + 8 * kh + v;
      h[(size_t)mr * NCLASS + ncol] = acc[v] + bb;
    }
  }
}

// z_out = alpha * h  (scatter base for one APPNP hop)
__global__ __launch_bounds__(256) void appnp_scale(
    float* __restrict__ zo, const float* __restrict__ h)
{
  const int i = blockIdx.x * 256 + threadIdx.x;
  if (i < N_NODES * NCLASS) zo[i] = 0.1f * h[i];
}

// z_out[dst] += (1-alpha) * val * z_in[src] ; 32 lanes cover one 40-wide edge row
__global__ __launch_bounds__(256) void appnp_scatter(
    const int* __restrict__ src, const int* __restrict__ dst,
    const float* __restrict__ val, const float* __restrict__ zin,
    float* __restrict__ zout)
{
  const int t = blockIdx.x * 256 + threadIdx.x;
  const int e = t >> 5;
  const int c = t & 31;
  if (e >= N_EDGES) return;
  const int s = src[e], d = dst[e];
  const float v = val[e] * 0.9f;
  const float* zi = zin  + (size_t)s * NCLASS;
  float*       zo = zout + (size_t)d * NCLASS;
  atomicAdd(&zo[c], v * zi[c]);
  if (c < NCLASS - 32) atomicAdd(&zo[c + 32], v * zi[c + 32]);
}

// log_softmax over 40-wide rows, one wave32 per row (safe in-place)
__global__ __launch_bounds__(256) void appnp_logsoftmax(
    const float* __restrict__ z, float* __restrict__ out)
{
  const int row  = blockIdx.x * 8 + (threadIdx.x >> 5);
  const int lane = threadIdx.x & 31;
  const float* zr = z + (size_t)row * NCLASS;
  const float a = zr[lane];
  const float b = (lane < 8) ? zr[lane + 32] : -3.402823466e38f;
  float m = fmaxf(a, b);
#pragma unroll
  for (int off = 16; off > 0; off >>= 1) m = fmaxf(m, __shfl_xor(m, off, 32));
  float s = __expf(a - m) + ((lane < 8) ? __expf(b - m) : 0.0f);
#pragma unroll
  for (int off = 16; off > 0; off >>= 1) s += __shfl_xor(s, off, 32);
  const float l = m + __logf(s);
  float* orow = out + (size_t)row * NCLASS;
  orow[lane] = a - l;
  if (lane < 8) orow[lane + 32] = b - l;
}

extern "C" void kernel_launch(void* const* d_in, const int* in_sizes, int n_in,
                              void* d_out, int out_size, void* d_ws, size_t ws_size,
                              hipStream_t stream) {
  const float* x    = (const float*)d_in[0];
  const int*   esrc = (const int*)  d_in[1];
  const int*   edst = (const int*)  d_in[2];
  const float* eval = (const float*)d_in[3];
  const float* W1   = (const float*)d_in[4];
  const float* b1   = (const float*)d_in[5];
  const float* W2   = (const float*)d_in[6];
  const float* b2   = (const float*)d_in[7];

  float* h  = (float*)d_ws;                          // N x 40  (16 MB)
  float* zA = h + (size_t)N_NODES * NCLASS;          // ping    (16 MB)
  float* zB = (float*)d_out;                         // pong = d_out

  appnp_mlp<<<N_NODES / 32, 256, 0, stream>>>(x, W1, b1, W2, b2, h);

  const float* zin = h;
  for (int hop = 0; hop < K_HOPS; ++hop) {
    float* zout = (hop & 1) ? zB : zA;
    appnp_scale<<<(N_NODES * NCLASS + 255) / 256, 256, 0, stream>>>(zout, h);
    appnp_scatter<<<(N_EDGES * 32) / 256, 256, 0, stream>>>(esrc, edst, eval, zin, zout);
    zin = zout;
  }
  // hop 9 (odd) landed in zB == d_out; log_softmax runs in place.
  appnp_logsoftmax<<<N_NODES / 8, 256, 0, stream>>>(zin, (float*)d_out);
}